// MyViT_16037407883837
// MI455X (gfx1250) — compile-verified
//
#include <hip/hip_runtime.h>

typedef __attribute__((ext_vector_type(2))) float v2f;
typedef __attribute__((ext_vector_type(8))) float v8f;

#define NPATCH 14
#define PS 16
#define CCH 3
#define HIMG 224
#define WIMG 224
#define KDIM 768   // input_d  = 3*16*16
#define NDIM 768   // hidden_d
#define ROWS_PER_WG 16
#define LDS_STRIDE 772   // 772 % 64 == 4 -> rows land 4 banks apart, conflict-free b64 reads
#define NROWS_TOTAL (64 * 196)   // 12544 patch rows

// One workgroup: 256 threads = 8 wave32s. Computes 16 output rows x all 768 cols.
__global__ __launch_bounds__(256)
void vit_embed_gemm(const float* __restrict__ img,
                    const float* __restrict__ Wt,     // [768 (h), 768 (d)] row-major
                    const float* __restrict__ bias,   // [768]
                    float* __restrict__ out)          // [64, 197, 768]
{
    __shared__ float As[ROWS_PER_WG * LDS_STRIDE];

    const int tid = threadIdx.x;
    const int gr0 = blockIdx.x * ROWS_PER_WG;   // base global patch-row

    // ---- Stage patchified A-tile into LDS ----------------------------------
    // 16 rows x 768 floats = 3072 float4 segments; 256 threads x 12 each.
    // Patch element d = c*256 + ph*16 + pw ; pw runs contiguous in memory,
    // so float4 along pw is fully coalesced per (c,ph) segment.
    #pragma unroll
    for (int it = 0; it < 12; ++it) {
        int idx = tid + it * 256;          // float4 index within tile
        int r   = idx / 192;               // tile row
        int f   = idx % 192;               // float4 within row
        int d   = f * 4;
        int c   = d >> 8;                  // d / 256
        int rem = d & 255;
        int ph  = rem >> 4;
        int pw  = rem & 15;
        int grow = gr0 + r;
        int b   = grow / 196;
        int p   = grow % 196;
        int pi  = p / NPATCH;
        int pj  = p % NPATCH;
        const float4 v = *reinterpret_cast<const float4*>(
            img + (((size_t)b * CCH + c) * HIMG + (pi * PS + ph)) * WIMG
                + pj * PS + pw);
        *reinterpret_cast<float4*>(&As[r * LDS_STRIDE + d]) = v;
    }
    __syncthreads();

    // ---- WMMA compute -------------------------------------------------------
    const int wave  = tid >> 5;
    const int lane  = tid & 31;
    const int lrow  = lane & 15;          // A row / B column inside 16x16 tile
    const int khalf = (lane >> 4) << 1;   // 0 or 2 : K sub-pair per lane half

    const float* Arow = &As[lrow * LDS_STRIDE + khalf];

    #pragma unroll 1
    for (int t = 0; t < 6; ++t) {         // 48 N-tiles / 8 waves = 6 per wave
        const int nt = wave + (t << 3);
        const int n0 = nt * 16;
        const float* Brow = Wt + (size_t)(n0 + lrow) * KDIM + khalf;

        v8f acc = {0.f, 0.f, 0.f, 0.f, 0.f, 0.f, 0.f, 0.f};

        #pragma unroll 4
        for (int k0 = 0; k0 < KDIM; k0 += 4) {
            // A fragment: lane holds A[lrow, k0+khalf .. +1]  (ds_load_b64)
            v2f a = *reinterpret_cast<const v2f*>(Arow + k0);
            // B fragment: lane holds B[k0+khalf .. +1, n0+lrow] = W[n0+lrow, k..]
            v2f bf = *reinterpret_cast<const v2f*>(Brow + k0);
            acc = __builtin_amdgcn_wmma_f32_16x16x4_f32(
                false, a, false, bf, (short)0, acc, false, false);
        }

        // C/D layout: lane l, VGPR j -> M = j + 8*(l/16), N = l%16
        const float bv = bias[n0 + lrow];
        #pragma unroll
        for (int j = 0; j < 8; ++j) {
            int m    = j + ((lane >> 4) << 3);
            int grow = gr0 + m;
            int b    = grow / 196;
            int p    = grow % 196;
            out[(((size_t)b * 197) + p + 1) * NDIM + n0 + lrow] = acc[j] + bv;
        }
    }
}

// Broadcast class token into row p=0 of every batch element.
__global__ __launch_bounds__(256)
void vit_cls(const float* __restrict__ cls, float* __restrict__ out)
{
    int idx = blockIdx.x * 256 + threadIdx.x;   // 0 .. 64*768-1
    int b = idx / NDIM;
    int h = idx % NDIM;
    out[(size_t)b * 197 * NDIM + h] = cls[h];
}

extern "C" void kernel_launch(void* const* d_in, const int* in_sizes, int n_in,
                              void* d_out, int out_size, void* d_ws, size_t ws_size,
                              hipStream_t stream)
{
    const float* img  = (const float*)d_in[0];  // [64,3,224,224]
    const float* Wt   = (const float*)d_in[1];  // [768,768]
    const float* bias = (const float*)d_in[2];  // [768]
    const float* cls  = (const float*)d_in[3];  // [1,768]
    float* out = (float*)d_out;                 // [64,197,768]

    vit_embed_gemm<<<NROWS_TOTAL / ROWS_PER_WG, 256, 0, stream>>>(img, Wt, bias, out);
    vit_cls<<<(64 * NDIM) / 256, 256, 0, stream>>>(cls, out);
}